// PointNetSetAbstraction_36197984371436
// MI455X (gfx1250) — compile-verified
//
#include <hip/hip_runtime.h>
#include <stdint.h>
#include <stddef.h>

#define BB 8
#define NN 8192
#define SS 1024
#define KG 32
#define RAD2 0.16f
#define EPSV 1e-5f

typedef __attribute__((ext_vector_type(2))) float v2f;
typedef __attribute__((ext_vector_type(8))) float v8f;

// ---------------------------------------------------------------------------
// 1) Farthest point sampling: one block per batch, points in registers.
//    Argmax per step: register-local best -> wave32 shfl_xor butterfly ->
//    32 candidates in LDS -> wave-0 butterfly. 3 barriers/step (was 11).
// ---------------------------------------------------------------------------
__global__ __launch_bounds__(1024) void fps_kernel(const float* __restrict__ xyz,
                                                   float* __restrict__ new_xyz) {
  __shared__ float cent[4];
  __shared__ float swv[32];
  __shared__ int   swi[32];
  __shared__ int   sfar;
  const int b = blockIdx.x;
  const int t = threadIdx.x;
  const int lane = t & 31;
  const int wv = t >> 5;
  const float* xb = xyz + (size_t)b * 3 * NN;
  float px[8], py[8], pz[8], dist[8];
#pragma unroll
  for (int i = 0; i < 8; ++i) {
    int n = i * 1024 + t;
    px[i] = xb[n];
    py[i] = xb[NN + n];
    pz[i] = xb[2 * NN + n];
    dist[i] = 1e10f;
  }
  int far = 0;
  for (int s = 0; s < SS; ++s) {
    if ((far & 1023) == t) {                    // owner broadcasts centroid
      int i = far >> 10;
      cent[0] = px[i]; cent[1] = py[i]; cent[2] = pz[i];
    }
    __syncthreads();
    float cx = cent[0], cy = cent[1], cz = cent[2];
    if (t == 0) {
      float* o = new_xyz + ((size_t)b * SS + s) * 3;
      o[0] = cx; o[1] = cy; o[2] = cz;
    }
    float bv = -1.0f;
    int   bi = 0x7fffffff;
#pragma unroll
    for (int i = 0; i < 8; ++i) {
      float dx = px[i] - cx, dy = py[i] - cy, dz = pz[i] - cz;
      float d = dx * dx + dy * dy + dz * dz;
      dist[i] = fminf(dist[i], d);
      int n = i * 1024 + t;
      if (dist[i] > bv || (dist[i] == bv && n < bi)) { bv = dist[i]; bi = n; }
    }
    // wave32 butterfly argmax (tie -> smaller index)
#pragma unroll
    for (int o = 16; o > 0; o >>= 1) {
      float ov = __shfl_xor(bv, o);
      int   oi = __shfl_xor(bi, o);
      if (ov > bv || (ov == bv && oi < bi)) { bv = ov; bi = oi; }
    }
    if (lane == 0) { swv[wv] = bv; swi[wv] = bi; }
    __syncthreads();
    if (t < 32) {                               // wave 0 reduces 32 candidates
      float v = swv[t];
      int   i2 = swi[t];
#pragma unroll
      for (int o = 16; o > 0; o >>= 1) {
        float ov = __shfl_xor(v, o);
        int   oi = __shfl_xor(i2, o);
        if (ov > v || (ov == v && oi < i2)) { v = ov; i2 = oi; }
      }
      if (t == 0) sfar = i2;
    }
    __syncthreads();
    far = sfar;
  }
}

// ---------------------------------------------------------------------------
// 2) Transpose points (B,64,N) -> pts_t (B,N,64), coalesced both directions.
// ---------------------------------------------------------------------------
__global__ __launch_bounds__(256) void transpose_pts_kernel(
    const float* __restrict__ points, float* __restrict__ pts_t) {
  __shared__ float tile[32][33];
  const int b = blockIdx.z, ct = blockIdx.y, ntile = blockIdx.x;
  const int tx = threadIdx.x & 31;
  const int ty = threadIdx.x >> 5;
  const float* src = points + (size_t)b * 64 * NN;
#pragma unroll
  for (int cc = ty; cc < 32; cc += 8)
    tile[cc][tx] = src[(size_t)(ct * 32 + cc) * NN + ntile * 32 + tx];
  __syncthreads();
  float* dst = pts_t + ((size_t)b * NN + ntile * 32) * 64;
#pragma unroll
  for (int nn = ty; nn < 32; nn += 8)
    dst[(size_t)nn * 64 + ct * 32 + tx] = tile[tx][nn];
}

// ---------------------------------------------------------------------------
// 3) Ball query + grouping: one wave32 per query (K==32==wave). Ballot/popc
//    gives the first-K in-radius indices in ascending order (reference sort
//    semantics). Builds padded 68-col feature rows [gnorm(3)|pts(64)|0].
// ---------------------------------------------------------------------------
__global__ __launch_bounds__(256) void group_kernel(
    const float* __restrict__ xyz, const float* __restrict__ pts_t,
    const float* __restrict__ new_xyz,
    float* __restrict__ out_idx, float* __restrict__ out_gnorm,
    float* __restrict__ out_g, float* __restrict__ out_mean,
    float* __restrict__ out_pnorm, float* __restrict__ feat) {
  __shared__ int sidx[8][KG];
  const int wid  = threadIdx.x >> 5;
  const int lane = threadIdx.x & 31;
  const int q = blockIdx.x * 8 + wid;        // global query (b*S + s)
  const int b = q / SS;
  const float* xb = xyz + (size_t)b * 3 * NN;
  const float cx = new_xyz[(size_t)q * 3 + 0];
  const float cy = new_xyz[(size_t)q * 3 + 1];
  const float cz = new_xyz[(size_t)q * 3 + 2];

  int cnt = 0;
  for (int n0 = 0; n0 < NN && cnt < KG; n0 += 32) {
    int n = n0 + lane;
    float dx = xb[n] - cx, dy = xb[NN + n] - cy, dz = xb[2 * NN + n] - cz;
    float d2 = dx * dx + dy * dy + dz * dz;
    bool inr = d2 <= RAD2;
    unsigned mask = (unsigned)__ballot(inr);
    if (inr) {
      int r = cnt + __popc(mask & ((1u << lane) - 1u));
      if (r < KG) sidx[wid][r] = n;
    }
    cnt += __popc(mask);
  }
  if (cnt > KG) cnt = KG;
  __syncthreads();

  int n = (lane < cnt) ? sidx[wid][lane] : sidx[wid][0];   // pad with first
  float gx = xb[n], gy = xb[NN + n], gz = xb[2 * NN + n];

  float mx = gx, my = gy, mz = gz;
#pragma unroll
  for (int o = 16; o > 0; o >>= 1) {
    mx += __shfl_xor(mx, o); my += __shfl_xor(my, o); mz += __shfl_xor(mz, o);
  }
  mx *= (1.0f / KG); my *= (1.0f / KG); mz *= (1.0f / KG);
  float sx = gx - mx, sy = gy - my, sz = gz - mz;
  float pn = sqrtf(sx * sx + sy * sy + sz * sz);
  float mxn = pn;
#pragma unroll
  for (int o = 16; o > 0; o >>= 1) mxn = fmaxf(mxn, __shfl_xor(mxn, o));
  float inv = 1.0f / mxn;
  float nx = sx * inv, ny = sy * inv, nz = sz * inv;

  size_t base = (size_t)q * KG + lane;
  out_idx[base] = (float)n;
  float* g3 = out_g + base * 3;      g3[0] = gx; g3[1] = gy; g3[2] = gz;
  float* q3 = out_gnorm + base * 3;  q3[0] = nx; q3[1] = ny; q3[2] = nz;
  if (lane == 0) {
    float* m3 = out_mean + (size_t)q * 3;
    m3[0] = mx; m3[1] = my; m3[2] = mz;
    out_pnorm[q] = mxn;
  }
  float* fr = feat + base * 68ull;
  fr[0] = nx; fr[1] = ny; fr[2] = nz;
  const float* pr = pts_t + ((size_t)b * NN + n) * 64;     // contiguous 256B
#pragma unroll 8
  for (int c = 0; c < 64; ++c) fr[3 + c] = pr[c];
  fr[67] = 0.0f;
}

// ---------------------------------------------------------------------------
// 4) Weight packing into WMMA B-fragment order:
//    Wp[nt][kstep][lane][2], zero-padded K -> branch-free b64 fragment loads.
// ---------------------------------------------------------------------------
__global__ __launch_bounds__(256) void pack_w_kernel(
    const float* __restrict__ W, float* __restrict__ Wp,
    int Kd, int Kpad, int Cout) {
  int o = blockIdx.x * 256 + threadIdx.x;
  int total = (Cout >> 4) * (Kpad >> 2) * 64;
  if (o >= total) return;
  int e = o & 1, lane = (o >> 1) & 31, rest = o >> 6;
  int ksteps = Kpad >> 2;
  int ks = rest % ksteps, nt = rest / ksteps;
  int k = ks * 4 + (lane >> 4) * 2 + e;
  int ncol = nt * 16 + (lane & 15);
  Wp[o] = (k < Kd) ? W[(size_t)k * Cout + ncol] : 0.0f;
}

// ---------------------------------------------------------------------------
// 5) FP32 WMMA GEMM with N-tile register blocking:
//    each wave owns 16 rows x (NT*16) cols; one A fragment feeds NT WMMAs.
//    Optional fused BN+ReLU on A operand; optional bias on output.
// ---------------------------------------------------------------------------
template <int NT, bool BN, bool BIAS>
__global__ __launch_bounds__(256) void gemm_wmma_kernel(
    const float* __restrict__ A, const float* __restrict__ Wp,
    const float* __restrict__ in_scale, const float* __restrict__ in_shift,
    const float* __restrict__ bias, float* __restrict__ out,
    int Kpad, int lda) {
  const int Cout = NT * 16;
  const int wid  = threadIdx.x >> 5;
  const int lane = threadIdx.x & 31;
  const int m0   = (blockIdx.x * 8 + wid) * 16;
  const int l15  = lane & 15;
  const int koff = (lane >> 4) * 2;          // ISA A-layout: lanes 16-31 hold K+2
  const float* arow = A + (size_t)(m0 + l15) * lda;
  const v2f* wfrag  = (const v2f*)Wp + lane;
  const int ksteps  = Kpad >> 2;

  v8f acc[NT];
#pragma unroll
  for (int nt = 0; nt < NT; ++nt) acc[nt] = (v8f){};

  for (int ks = 0; ks < ksteps; ++ks) {
    const int ka = ks * 4 + koff;
    v2f a = *(const v2f*)(arow + ka);
    if (BN) {                                 // fused BN + ReLU on input
      v2f sc = *(const v2f*)(in_scale + ka);
      v2f sh = *(const v2f*)(in_shift + ka);
      a.x = fmaxf(a.x * sc.x + sh.x, 0.0f);
      a.y = fmaxf(a.y * sc.y + sh.y, 0.0f);
    }
    __builtin_prefetch(arow + ka + 16, 0, 1); // stream A ahead
#pragma unroll
    for (int nt = 0; nt < NT; ++nt) {
      v2f wv = wfrag[(size_t)(nt * ksteps + ks) * 32];
      acc[nt] = __builtin_amdgcn_wmma_f32_16x16x4_f32(false, a, false, wv,
                                                      (short)0, acc[nt],
                                                      false, false);
    }
  }

  const int mhi = (lane >> 4) * 8;
#pragma unroll
  for (int nt = 0; nt < NT; ++nt) {
    const int ncol = nt * 16 + l15;
    float bv = BIAS ? bias[ncol] : 0.0f;
#pragma unroll
    for (int r = 0; r < 8; ++r)
      out[(size_t)(m0 + mhi + r) * Cout + ncol] = acc[nt][r] + bv;
  }
}

// ---------------------------------------------------------------------------
// 6) BN statistics: one block per channel; folds (mean,var,gamma,beta) into
//    scale/shift consumed by the next GEMM's operand load.
// ---------------------------------------------------------------------------
__global__ __launch_bounds__(256) void stats_kernel(
    const float* __restrict__ x, int M, int C,
    const float* __restrict__ g, const float* __restrict__ be,
    float* __restrict__ scale, float* __restrict__ shift) {
  __shared__ float ssum[256], ssq[256];
  const int c = blockIdx.x, t = threadIdx.x;
  float s = 0.0f, q = 0.0f;
  for (int i = t; i < M; i += 256) {
    float v = x[(size_t)i * C + c];
    s += v; q += v * v;
  }
  ssum[t] = s; ssq[t] = q;
  __syncthreads();
  for (int o = 128; o > 0; o >>= 1) {
    if (t < o) { ssum[t] += ssum[t + o]; ssq[t] += ssq[t + o]; }
    __syncthreads();
  }
  if (t == 0) {
    float mean = ssum[0] / (float)M;
    float var  = ssq[0] / (float)M - mean * mean;
    float sc = g[c] * rsqrtf(var + EPSV);
    scale[c] = sc;
    shift[c] = be[c] - mean * sc;
  }
}

// ---------------------------------------------------------------------------
// 7) K-max pool with fused BN+ReLU of layer 3.
// ---------------------------------------------------------------------------
__global__ __launch_bounds__(256) void pool_kernel(
    const float* __restrict__ act3, const float* __restrict__ scale,
    const float* __restrict__ shift, float* __restrict__ pooled) {
  int i = blockIdx.x * 256 + threadIdx.x;   // over B*S*128
  int c = i & 127, s = i >> 7;
  float sc = scale[c], sh = shift[c];
  const float* p = act3 + ((size_t)s * KG) * 128 + c;
  float m = 0.0f;                            // ReLU output is >= 0
#pragma unroll 8
  for (int k = 0; k < KG; ++k)
    m = fmaxf(m, fmaxf(p[(size_t)k * 128] * sc + sh, 0.0f));
  pooled[i] = m;
}

// ---------------------------------------------------------------------------
extern "C" void kernel_launch(void* const* d_in, const int* in_sizes, int n_in,
                              void* d_out, int out_size, void* d_ws, size_t ws_size,
                              hipStream_t stream) {
  const float* xyz    = (const float*)d_in[0];
  const float* points = (const float*)d_in[1];
  const float* w1_0 = (const float*)d_in[2];
  const float* g1_0 = (const float*)d_in[4];
  const float* be1_0= (const float*)d_in[5];
  const float* w1_1 = (const float*)d_in[6];
  const float* g1_1 = (const float*)d_in[8];
  const float* be1_1= (const float*)d_in[9];
  const float* w1_2 = (const float*)d_in[10];
  const float* g1_2 = (const float*)d_in[12];
  const float* be1_2= (const float*)d_in[13];
  const float* w2_0 = (const float*)d_in[14];
  const float* g2_0 = (const float*)d_in[16];
  const float* be2_0= (const float*)d_in[17];
  const float* w2_1 = (const float*)d_in[18];
  const float* b2_1 = (const float*)d_in[19];

  // output layout (floats, reference return order)
  float* out = (float*)d_out;
  float* o_newxyz = out;                                  // 8*1024*3
  float* o_newpts = out + 24576;                          // 8*1024*256
  float* o_idx    = out + 24576 + 2097152;                // 8*1024*32
  float* o_gnorm  = o_idx + 262144;                       // 8*1024*32*3
  float* o_g      = o_gnorm + 786432;                     // 8*1024*32*3
  float* o_mean   = o_g + 786432;                         // 8*1024*1*3
  float* o_pnorm  = o_mean + 24576;                       // 8*1024

  // workspace layout (phase-reused)
  char* ws = (char*)d_ws;
  const int M1 = BB * SS * KG;                            // 262144
  const int M2 = BB * SS;                                 // 8192
  float* feat  = (float*)(ws);                            // [M1][68]  71.3MB
  float* act1  = (float*)(ws + 71303168ull);              // [M1][64]  67.1MB
  float* act2  = (float*)(ws + 138412032ull);             // [M1][64]  67.1MB
  float* pts_t = (float*)(ws + 138412032ull);             // [B][N][64] 16MB (dead before act2 written)
  float* act3  = (float*)(ws);                            // [M1][128] reuses feat+act1 (dead)
  float* pooled= (float*)(ws + 134217728ull);             // [M2][128] in dead act1 tail
  float* act4  = (float*)(ws + 138412032ull);             // [M2][128] reuses act2 (dead)
  float* prm   = (float*)(ws + 205520896ull);             // scale/shift arrays
  float* sc1 = prm,        *sh1 = prm + 64;
  float* sc2 = prm + 128,  *sh2 = prm + 192;
  float* sc3 = prm + 256,  *sh3 = prm + 384;
  float* sc4 = prm + 512,  *sh4 = prm + 640;
  float* wpb = (float*)(ws + 205524992ull);               // packed weights
  float* wp1 = wpb;              // 68*64   = 4352
  float* wp2 = wp1 + 4352;       // 64*64   = 4096
  float* wp3 = wp2 + 4096;       // 64*128  = 8192
  float* wp4 = wp3 + 8192;       // 128*128 = 16384
  float* wp5 = wp4 + 16384;      // 128*256 = 32768

  // weight packing (independent, cheap)
  pack_w_kernel<<<17, 256, 0, stream>>>(w1_0, wp1, 67, 68, 64);
  pack_w_kernel<<<16, 256, 0, stream>>>(w1_1, wp2, 64, 64, 64);
  pack_w_kernel<<<32, 256, 0, stream>>>(w1_2, wp3, 64, 64, 128);
  pack_w_kernel<<<64, 256, 0, stream>>>(w2_0, wp4, 128, 128, 128);
  pack_w_kernel<<<128, 256, 0, stream>>>(w2_1, wp5, 128, 128, 256);

  // points transpose for coalesced channel gather
  transpose_pts_kernel<<<dim3(NN / 32, 2, BB), 256, 0, stream>>>(points, pts_t);

  // 1) FPS -> new_xyz
  fps_kernel<<<BB, 1024, 0, stream>>>(xyz, o_newxyz);

  // 2) ball query + grouping + feature assembly
  group_kernel<<<(BB * SS) / 8, 256, 0, stream>>>(
      xyz, pts_t, o_newxyz, o_idx, o_gnorm, o_g, o_mean, o_pnorm, feat);

  // 3) MLP1 layer 0: [M1,67(pad 68)] x [68,64]
  gemm_wmma_kernel<4, false, false><<<M1 / 128, 256, 0, stream>>>(
      feat, wp1, nullptr, nullptr, nullptr, act1, 68, 68);
  stats_kernel<<<64, 256, 0, stream>>>(act1, M1, 64, g1_0, be1_0, sc1, sh1);

  // 4) MLP1 layer 1 (BN+ReLU fused into A-load)
  gemm_wmma_kernel<4, true, false><<<M1 / 128, 256, 0, stream>>>(
      act1, wp2, sc1, sh1, nullptr, act2, 64, 64);
  stats_kernel<<<64, 256, 0, stream>>>(act2, M1, 64, g1_1, be1_1, sc2, sh2);

  // 5) MLP1 layer 2
  gemm_wmma_kernel<8, true, false><<<M1 / 128, 256, 0, stream>>>(
      act2, wp3, sc2, sh2, nullptr, act3, 64, 64);
  stats_kernel<<<128, 256, 0, stream>>>(act3, M1, 128, g1_2, be1_2, sc3, sh3);

  // 6) BN+ReLU + max over K
  pool_kernel<<<(M2 * 128) / 256, 256, 0, stream>>>(act3, sc3, sh3, pooled);

  // 7) MLP2 layer 0
  gemm_wmma_kernel<8, false, false><<<M2 / 128, 256, 0, stream>>>(
      pooled, wp4, nullptr, nullptr, nullptr, act4, 128, 128);
  stats_kernel<<<128, 256, 0, stream>>>(act4, M2, 128, g2_0, be2_0, sc4, sh4);

  // 8) MLP2 layer 1 (+bias, no BN) -> new_points
  gemm_wmma_kernel<16, true, true><<<M2 / 128, 256, 0, stream>>>(
      act4, wp5, sc4, sh4, b2_1, o_newpts, 128, 128);
}